// CAMEncoder3_49770081026694
// MI455X (gfx1250) — compile-verified
//
#include <hip/hip_runtime.h>

// GATv2 encoder for MI455X (gfx1250, wave32).
// Dense transforms use V_WMMA_F32_16X16X4_F32 (exact fp32 matrix path).
// Edge phase: wave-per-edge coalesced gathers + L2-resident float atomics.

typedef __attribute__((ext_vector_type(2))) float v2f;
typedef __attribute__((ext_vector_type(8))) float v8f;

#define LRELU_SLOPE 0.15f

// ---- monotone float<->int key for atomicMax-based segment max ----
__device__ __forceinline__ int f32key(float f) {
  int i = __float_as_int(f);
  return i >= 0 ? i : (i ^ 0x7fffffff);
}
__device__ __forceinline__ float keyf32(int k) {
  return __int_as_float(k >= 0 ? k : (k ^ 0x7fffffff));
}

// ---------------- fills ----------------
__global__ void fill_f32(float* __restrict__ p, float v, long long n) {
  long long t = (long long)blockIdx.x * blockDim.x + threadIdx.x;
  if (t < n) p[t] = v;
}
__global__ void fill_i32(int* __restrict__ p, int v, long long n) {
  long long t = (long long)blockIdx.x * blockDim.x + threadIdx.x;
  if (t < n) p[t] = v;
}

// ---------------- WMMA GEMM:  Y[M,NC] = X[M,K] @ W[K,NC] + bias ----------------
// One wave computes a 16-row x 64-col strip (4 WMMA 16x16 tiles, A reused).
// M must be a multiple of 16 (50000 = 3125*16), NC a multiple of 64.
template <int K, int NC>
__global__ __launch_bounds__(256) void gemm_wmma(const float* __restrict__ X,
                                                 const float* __restrict__ W,
                                                 const float* __restrict__ bias,
                                                 float* __restrict__ Y, int M) {
  constexpr int NCG = NC / 64;
  int wave = blockIdx.x * 8 + (threadIdx.x >> 5);
  int ntiles = (M >> 4) * NCG;
  if (wave >= ntiles) return;  // wave-uniform exit: EXEC stays all-ones for WMMA
  int rowt = wave / NCG;
  int colg = wave % NCG;
  int lane = threadIdx.x & 31;
  int khi = lane >> 4;     // K-half select (A) / row pair select (B)
  int mn = lane & 15;      // M index for A, N index for B/C/D
  int row = (rowt << 4) + mn;
  int coln = (colg << 6) + mn;

  const float* xrow = X + (size_t)row * K + 2 * khi;
  v8f a0 = {}, a1 = {}, a2 = {}, a3 = {};

#pragma unroll
  for (int k = 0; k < K; k += 4) {
    // A fragment: lane holds X[row][k+2*khi .. k+2*khi+1] (8B aligned: K even, k%4==0)
    v2f af = *reinterpret_cast<const v2f*>(xrow + k);
    const float* w0 = W + (size_t)(k + 2 * khi) * NC + coln;
    const float* w1 = w0 + NC;
    v2f b0 = {w0[0],  w1[0]};
    v2f b1 = {w0[16], w1[16]};
    v2f b2 = {w0[32], w1[32]};
    v2f b3 = {w0[48], w1[48]};
    a0 = __builtin_amdgcn_wmma_f32_16x16x4_f32(false, af, false, b0, (short)0, a0, false, false);
    a1 = __builtin_amdgcn_wmma_f32_16x16x4_f32(false, af, false, b1, (short)0, a1, false, false);
    a2 = __builtin_amdgcn_wmma_f32_16x16x4_f32(false, af, false, b2, (short)0, a2, false, false);
    a3 = __builtin_amdgcn_wmma_f32_16x16x4_f32(false, af, false, b3, (short)0, a3, false, false);
  }

  float bv0 = bias[coln];
  float bv1 = bias[coln + 16];
  float bv2 = bias[coln + 32];
  float bv3 = bias[coln + 48];
  int rbase = (rowt << 4) + (khi << 3);  // D: vgpr r -> row r (lanes 0-15), r+8 (lanes 16-31)
#pragma unroll
  for (int r = 0; r < 8; ++r) {
    float* yp = Y + (size_t)(rbase + r) * NC + coln;
    yp[0]  = a0[r] + bv0;
    yp[16] = a1[r] + bv1;
    yp[32] = a2[r] + bv2;
    yp[48] = a3[r] + bv3;
  }
}

// ---------------- Edge kernels (wave per edge) ----------------
// logits[e,h] = sum_c att[h,c] * leakyrelu(xl[src][h,c] + xr[dst][h,c]);  seg-max via atomicMax
__global__ __launch_bounds__(256) void edge_logits(const float* __restrict__ xl,
                                                   const float* __restrict__ xr,
                                                   const float* __restrict__ att,
                                                   const int* __restrict__ src,
                                                   const int* __restrict__ dst,
                                                   float* __restrict__ logit,
                                                   int* __restrict__ mkey,
                                                   int E, int HC, int C) {
  int e = blockIdx.x * 8 + (threadIdx.x >> 5);
  if (e >= E) return;
  int lane = threadIdx.x & 31;
  int s = src[e], d = dst[e];
  const float* pl = xl + (size_t)s * HC;
  const float* pr = xr + (size_t)d * HC;
  float p0 = 0.f, p1 = 0.f;
  for (int idx = lane; idx < HC; idx += 32) {
    float v = pl[idx] + pr[idx];
    v = v > 0.f ? v : LRELU_SLOPE * v;
    float c = v * att[idx];
    if (idx < C) p0 += c; else p1 += c;
  }
#pragma unroll
  for (int off = 16; off > 0; off >>= 1) {
    p0 += __shfl_xor(p0, off, 32);
    p1 += __shfl_xor(p1, off, 32);
  }
  if (lane == 0) {
    logit[(size_t)e * 2 + 0] = p0;
    logit[(size_t)e * 2 + 1] = p1;
    atomicMax(mkey + (size_t)d * 2 + 0, f32key(p0));
    atomicMax(mkey + (size_t)d * 2 + 1, f32key(p1));
  }
}

// ex = exp(logit - m[dst]) (in place), denom[dst,h] += ex
__global__ void edge_exp(float* __restrict__ logit, const int* __restrict__ dst,
                         const int* __restrict__ mkey, float* __restrict__ denom,
                         long long EH) {
  long long t = (long long)blockIdx.x * blockDim.x + threadIdx.x;
  if (t >= EH) return;
  int e = (int)(t >> 1);
  int h = (int)(t & 1);
  int d = dst[e];
  float m = keyf32(mkey[(size_t)d * 2 + h]);  // always finite: this edge contributed
  float ex = __expf(logit[t] - m);
  logit[t] = ex;
  atomicAdd(denom + (size_t)d * 2 + h, ex);
}

// acc[dst,h,c] += (ex / denom[dst,h]) * xl[src,h,c]
__global__ __launch_bounds__(256) void edge_scatter(const float* __restrict__ xl,
                                                    const float* __restrict__ ex,
                                                    const float* __restrict__ denom,
                                                    const int* __restrict__ src,
                                                    const int* __restrict__ dst,
                                                    float* __restrict__ acc,
                                                    int E, int HC, int C) {
  int e = blockIdx.x * 8 + (threadIdx.x >> 5);
  if (e >= E) return;
  int lane = threadIdx.x & 31;
  int s = src[e], d = dst[e];
  float a0 = ex[(size_t)e * 2 + 0] / (denom[(size_t)d * 2 + 0] + 1e-16f);
  float a1 = ex[(size_t)e * 2 + 1] / (denom[(size_t)d * 2 + 1] + 1e-16f);
  const float* pl = xl + (size_t)s * HC;
  float* pa = acc + (size_t)d * HC;
  for (int idx = lane; idx < HC; idx += 32) {
    float al = (idx < C) ? a0 : a1;
    atomicAdd(pa + idx, al * pl[idx]);
  }
}

// ---------------- node kernels ----------------
// head mean + bias; mode 0: relu+write, 1: write, 2: add
__global__ void node_finalize(const float* __restrict__ acc, const float* __restrict__ bias,
                              float* __restrict__ dest, int ldd, int coff, int C, int mode,
                              int n) {
  long long t = (long long)blockIdx.x * blockDim.x + threadIdx.x;
  if (t >= (long long)n * C) return;
  int i = (int)(t / C);
  int c = (int)(t % C);
  float v = (acc[(size_t)i * 2 * C + c] + acc[(size_t)i * 2 * C + C + c]) * 0.5f + bias[c];
  float* dp = dest + (size_t)i * ldd + coff + c;
  if (mode == 0)      *dp = v > 0.f ? v : 0.f;
  else if (mode == 1) *dp = v;
  else                *dp += v;
}

__global__ void copy_x(const float* __restrict__ x, float* __restrict__ hbuf, int n) {
  long long t = (long long)blockIdx.x * blockDim.x + threadIdx.x;
  if (t >= (long long)n * 64) return;
  int i = (int)(t >> 6);
  int c = (int)(t & 63);
  hbuf[(size_t)i * 192 + c] = x[t];
}

// out = [x | relu(0.5 * accumulated(h2j+h2m))]
__global__ void final_out(const float* __restrict__ x, float* __restrict__ out, int n) {
  long long t = (long long)blockIdx.x * blockDim.x + threadIdx.x;
  if (t >= (long long)n * 192) return;
  int i = (int)(t / 192);
  int c = (int)(t % 192);
  if (c < 64) {
    out[t] = x[(size_t)i * 64 + c];
  } else {
    float v = out[t] * 0.5f;
    out[t] = v > 0.f ? v : 0.f;
  }
}

// ---------------- host-side pipeline ----------------
static inline int divup(long long a, long long b) { return (int)((a + b - 1) / b); }

template <int K, int NC>
static void run_conv(const float* X, const float* const* P,  // Wl, bl, Wr, br, att, b
                     const int* src, const int* dst, int Nn, int Ecnt,
                     float* xl, float* xr, float* logit, int* mkey, float* denom, float* acc,
                     float* dest, int ldd, int coff, int mode, hipStream_t stream) {
  constexpr int C = NC / 2;
  int waves = (Nn / 16) * (NC / 64);
  gemm_wmma<K, NC><<<divup(waves, 8), 256, 0, stream>>>(X, P[0], P[1], xl, Nn);
  gemm_wmma<K, NC><<<divup(waves, 8), 256, 0, stream>>>(X, P[2], P[3], xr, Nn);
  fill_i32<<<divup((long long)Nn * 2, 256), 256, 0, stream>>>(mkey, (int)0x80000000,
                                                              (long long)Nn * 2);
  fill_f32<<<divup((long long)Nn * 2, 256), 256, 0, stream>>>(denom, 0.f, (long long)Nn * 2);
  fill_f32<<<divup((long long)Nn * NC, 256), 256, 0, stream>>>(acc, 0.f, (long long)Nn * NC);
  edge_logits<<<divup(Ecnt, 8), 256, 0, stream>>>(xl, xr, P[4], src, dst, logit, mkey,
                                                  Ecnt, NC, C);
  edge_exp<<<divup((long long)Ecnt * 2, 256), 256, 0, stream>>>(logit, dst, mkey, denom,
                                                                (long long)Ecnt * 2);
  edge_scatter<<<divup(Ecnt, 8), 256, 0, stream>>>(xl, logit, denom, src, dst, acc,
                                                   Ecnt, NC, C);
  node_finalize<<<divup((long long)Nn * C, 256), 256, 0, stream>>>(acc, P[5], dest, ldd, coff,
                                                                   C, mode, Nn);
}

extern "C" void kernel_launch(void* const* d_in, const int* in_sizes, int n_in,
                              void* d_out, int out_size, void* d_ws, size_t ws_size,
                              hipStream_t stream) {
  (void)n_in; (void)out_size; (void)ws_size;
  const float* x  = (const float*)d_in[0];
  const int*   je = (const int*)d_in[1];
  const int*   me = (const int*)d_in[2];
  const float* P[24];
  for (int i = 0; i < 24; ++i) P[i] = (const float*)d_in[3 + i];
  // P[0..5]=jg1, P[6..11]=mg1, P[12..17]=jg2, P[18..23]=mg2 (Wl,bl,Wr,br,att,b each)

  int Nn   = in_sizes[0] / 64;   // 50000 (multiple of 16)
  int Ecnt = in_sizes[1] / 2;    // 500000

  // workspace arena (floats): xl, xr, acc sized for layer2 width (256)
  float* ws    = (float*)d_ws;
  float* xl    = ws;
  float* xr    = xl   + (size_t)Nn * 256;
  float* acc   = xr   + (size_t)Nn * 256;
  float* hbuf  = acc  + (size_t)Nn * 256;   // [N,192] = [x | h1j | h1m]
  float* logit = hbuf + (size_t)Nn * 192;   // [E,2] logits, then exp in place
  float* denom = logit + (size_t)Ecnt * 2;  // [N,2]
  int*   mkey  = (int*)(denom + (size_t)Nn * 2);  // [N,2] encoded seg-max
  float* out   = (float*)d_out;

  const int* jsrc = je, *jdst = je + Ecnt;
  const int* msrc = me, *mdst = me + Ecnt;

  // hbuf[:, 0:64] = x (layer-2 input concat)
  copy_x<<<divup((long long)Nn * 64, 256), 256, 0, stream>>>(x, hbuf, Nn);

  // layer 1: h1j, h1m with relu into hbuf cols [64:128), [128:192)
  run_conv<64, 128>(x, P + 0, jsrc, jdst, Nn, Ecnt, xl, xr, logit, mkey, denom, acc,
                    hbuf, 192, 64, /*relu*/0, stream);
  run_conv<64, 128>(x, P + 6, msrc, mdst, Nn, Ecnt, xl, xr, logit, mkey, denom, acc,
                    hbuf, 192, 128, /*relu*/0, stream);

  // layer 2: write h2j into d_out cols [64:192), add h2m
  run_conv<192, 256>(hbuf, P + 12, jsrc, jdst, Nn, Ecnt, xl, xr, logit, mkey, denom, acc,
                     out, 192, 64, /*write*/1, stream);
  run_conv<192, 256>(hbuf, P + 18, msrc, mdst, Nn, Ecnt, xl, xr, logit, mkey, denom, acc,
                     out, 192, 64, /*add*/2, stream);

  // out = [x | relu(0.5*(h2j+h2m))]
  final_out<<<divup((long long)Nn * 192, 256), 256, 0, stream>>>(x, out, Nn);
}